// HybridGNNModel_27685359190657
// MI455X (gfx1250) — compile-verified
//
#include <hip/hip_runtime.h>
#include <hip/hip_bf16.h>
#include <cstdint>

#define NN      131072
#define NE      524288
#define NG      4096
#define HID     128
#define HEADS   4
#define DHD     32
#define DESC    544
#define RH      104
#define NLAY    3
#define E2TOT   (NE + NN)
#define KT      128                 // K chunk staged in LDS per iteration
#define LDS_STRIDE (KT + 2)         // 130 floats: bank-conflict-free row stride

static constexpr float INV_BN = 0.9999950000374997f; // (1+1e-5)^-0.5

typedef __attribute__((ext_vector_type(2))) float v2f;
typedef __attribute__((ext_vector_type(8))) float v8f;
typedef __attribute__((ext_vector_type(4))) unsigned v4u;
typedef __attribute__((ext_vector_type(4))) int v4i;
typedef __attribute__((ext_vector_type(8))) int v8i;

// ---------- helpers ----------
__device__ __forceinline__ unsigned enc_f(float f) {
  unsigned u = __float_as_uint(f);
  return (u & 0x80000000u) ? ~u : (u | 0x80000000u);
}
__device__ __forceinline__ float dec_f(unsigned u) {
  return (u & 0x80000000u) ? __uint_as_float(u ^ 0x80000000u) : __uint_as_float(~u);
}

// ---------- node embedding: h = [emb_node[x_idx], x_feat @ w_nf.T + b_nf] ----------
__global__ void k_embed(const int* __restrict__ x_idx, const float* __restrict__ x_feat,
                        const float* __restrict__ emb_node, const float* __restrict__ w_nf,
                        const float* __restrict__ b_nf, float* __restrict__ h) {
  size_t idx = (size_t)blockIdx.x * blockDim.x + threadIdx.x;
  if (idx >= (size_t)NN * HID) return;
  int n = (int)(idx >> 7), c = (int)(idx & 127);
  float v;
  if (c < 64) {
    v = emb_node[x_idx[n] * 64 + c];
  } else {
    int j = c - 64;
    float acc = b_nf[j];
    const float* xf = x_feat + (size_t)n * 8;
    const float* w  = w_nf + (size_t)j * 8;
#pragma unroll
    for (int k = 0; k < 8; ++k) acc += xf[k] * w[k];
    v = acc;
  }
  h[idx] = v;
}

// ---------- self-loop attr: deg + segment_sum(e_attr,dst) ----------
__global__ void k_deg(const int* __restrict__ dst, float* __restrict__ deg) {
  int e = blockIdx.x * blockDim.x + threadIdx.x;
  if (e < NE) atomicAdd(&deg[dst[e]], 1.0f);
}
__global__ void k_loop_sum(const int* __restrict__ dst, const int* __restrict__ eai,
                           const float* __restrict__ emb_edge, float* __restrict__ lsum) {
  size_t idx = (size_t)blockIdx.x * blockDim.x + threadIdx.x;
  if (idx >= (size_t)NE * HID) return;
  int e = (int)(idx >> 7), c = (int)(idx & 127);
  atomicAdd(&lsum[(size_t)dst[e] * HID + c], emb_edge[(size_t)eai[e] * HID + c]);
}
__global__ void k_loop_div(const float* __restrict__ deg, float* __restrict__ lattr) {
  size_t idx = (size_t)blockIdx.x * blockDim.x + threadIdx.x;
  if (idx >= (size_t)NN * HID) return;
  lattr[idx] /= fmaxf(deg[idx >> 7], 1.0f);
}

// ---------- TDM: stage 128 rows x kt cols of W (row stride K) into LDS ----------
// Padded layout: LDS row r at byte offset lds_base + r*LDS_STRIDE*4.
// kt must be a power of two (128 or 32 here); pad_interval code = log2(kt)-1,
// pad_amount code = (LDS_STRIDE - kt) - 1.
__device__ __forceinline__ void tdm_stage_w(const float* gsrc, unsigned lds_base,
                                            int kt, int K) {
  unsigned long long ga = (unsigned long long)(const char*)gsrc;
  v4u g0;
  g0.x = 1u;                                         // count=1, no gather
  g0.y = lds_base;                                   // lds_addr (bytes)
  g0.z = (unsigned)(ga & 0xFFFFFFFFu);               // global_addr[31:0]
  g0.w = (unsigned)((ga >> 32) & 0x01FFFFFFu) | 0x80000000u; // addr[56:32] | type=2

  unsigned ic = (unsigned)(31 - __builtin_clz((unsigned)kt)) - 1u; // interval code
  unsigned ac = (unsigned)(LDS_STRIDE - kt - 1);                   // amount code
  v8i g1;
  g1[0] = (int)((2u << 16) | (1u << 20) | (ic << 22) | (ac << 25)); // 4B elems, pad on
  g1[1] = (int)(((unsigned)kt & 0xFFFFu) << 16);     // tensor_dim0[15:0] (=kt)
  g1[2] = (int)(((unsigned)kt >> 16) | (128u << 16)); // dim0 hi | tensor_dim1=128 lo
  g1[3] = (int)(((unsigned)kt & 0xFFFFu) << 16);     // dim1 hi=0 | tile_dim0=kt
  g1[4] = (int)128;                                  // tile_dim1=128 rows, tile_dim2=0
  g1[5] = (int)(unsigned)K;                          // tensor_dim0_stride[31:0]
  g1[6] = 0;                                         // stride hi, dim1_stride lo
  g1[7] = 0;
  v4i gz = {0, 0, 0, 0};
#if defined(__clang_major__) && (__clang_major__ >= 23)
  v8i gz8 = {0, 0, 0, 0, 0, 0, 0, 0};
  __builtin_amdgcn_tensor_load_to_lds(g0, g1, gz, gz, gz8, 0);
#else
  __builtin_amdgcn_tensor_load_to_lds(g0, g1, gz, gz, 0);
#endif
}

// ---------- WMMA fp32 GEMM: Out[M,O] = epi( X[M,K] @ W[O,K]^T ) ----------
// block = 128 threads (4 waves); each wave: 16 rows x 128 cols (8 WMMA tiles).
// grid.x = M/64, grid.y = O/128. K % 4 == 0; K chunks of KT are powers of two.
// W slab (128 rows x kt) staged in LDS by the Tensor Data Mover.
__global__ void k_gemm_wmma(const float* __restrict__ X, const float* __restrict__ W,
                            float* __restrict__ Out, const float* __restrict__ bias,
                            const float* __restrict__ gamma, const float* __restrict__ beta,
                            int K, int O, int relu) {
  __shared__ float wlds[128 * LDS_STRIDE];
  const int lane = threadIdx.x & 31;
  const int wave = threadIdx.x >> 5;
  const int m0 = blockIdx.x * 64 + wave * 16;
  const int n0 = blockIdx.y * 128;
  const int lm = lane & 15;
  const int koff = (lane >> 4) << 1;   // lanes 0-15 -> K+{0,1}; lanes 16-31 -> K+{2,3}
  const v8f vzero = {0.f, 0.f, 0.f, 0.f, 0.f, 0.f, 0.f, 0.f};
  v8f acc[8];
#pragma unroll
  for (int t = 0; t < 8; ++t) acc[t] = vzero;

  const unsigned lds_base = (unsigned)(unsigned long long)(void*)&wlds[0];
  const float* xrow = X + (size_t)(m0 + lm) * K + koff;

  for (int k0 = 0; k0 < K; k0 += KT) {
    const int kt = (K - k0 < KT) ? (K - k0) : KT;
    if (threadIdx.x < 32) {
      tdm_stage_w(W + (size_t)n0 * K + k0, lds_base, kt, K);
      __builtin_amdgcn_s_wait_tensorcnt(0);
    }
    __syncthreads();
#pragma unroll 4
    for (int k = 0; k < kt; k += 4) {
      v2f a = *(const v2f*)(xrow + k0 + k);
      v2f b[8];
#pragma unroll
      for (int t = 0; t < 8; ++t)
        b[t] = *(const v2f*)&wlds[(t * 16 + lm) * LDS_STRIDE + k + koff];
#pragma unroll
      for (int t = 0; t < 8; ++t)
        acc[t] = __builtin_amdgcn_wmma_f32_16x16x4_f32(false, a, false, b[t],
                                                       (short)0, acc[t], false, false);
    }
    __syncthreads();
  }

  const int mrow = m0 + ((lane >> 4) << 3);  // lanes 16-31 hold M rows +8
#pragma unroll
  for (int t = 0; t < 8; ++t) {
    int nn = n0 + t * 16 + lm;
    float bi = bias ? bias[nn] : 0.f;
    float g  = gamma ? gamma[nn] * INV_BN : 1.f;
    float be = gamma ? beta[nn] : 0.f;
#pragma unroll
    for (int j = 0; j < 8; ++j) {
      float v = (acc[t][j] + bi) * g + be;
      if (relu) v = fmaxf(v, 0.f);
      Out[(size_t)(mrow + j) * O + nn] = v;
    }
  }
}

// ---------- attention scalars per node: a_s, a_d ----------
__global__ void k_attn_nodes(const float* __restrict__ xh, const float* __restrict__ att_src,
                             const float* __restrict__ att_dst,
                             float* __restrict__ a_s, float* __restrict__ a_d) {
  int idx = blockIdx.x * blockDim.x + threadIdx.x;
  if (idx >= NN * HEADS) return;
  int n = idx >> 2, hh = idx & 3;
  const float* x  = xh + (size_t)n * HID + hh * DHD;
  const float* as = att_src + hh * DHD;
  const float* ad = att_dst + hh * DHD;
  float s = 0.f, d = 0.f;
#pragma unroll
  for (int k = 0; k < DHD; ++k) { float xv = x[k]; s += xv * as[k]; d += xv * ad[k]; }
  a_s[idx] = s; a_d[idx] = d;
}

// ---------- fold w_edge through att_edge: v[h,k] = sum_d w_edge[h*32+d,k]*att_edge[h,d] ----------
__global__ void k_edge_vec(const float* __restrict__ w_edge, const float* __restrict__ att_edge,
                           float* __restrict__ v) {
  int idx = blockIdx.x * blockDim.x + threadIdx.x;
  if (idx >= HEADS * HID) return;
  int hh = idx >> 7, k = idx & 127;
  float acc = 0.f;
  for (int d = 0; d < DHD; ++d)
    acc += w_edge[(size_t)(hh * DHD + d) * HID + k] * att_edge[hh * DHD + d];
  v[idx] = acc;
}

// ---------- alpha pre-activation + segment max (one thread per edge incl. self loops) ----------
__global__ void k_alpha_max(const int* __restrict__ srcA, const int* __restrict__ dstA,
                            const int* __restrict__ eai, const float* __restrict__ emb_edge,
                            const float* __restrict__ lattr, const float* __restrict__ vvec,
                            const float* __restrict__ a_s, const float* __restrict__ a_d,
                            float* __restrict__ pre, unsigned* __restrict__ m_u) {
  int e = blockIdx.x * blockDim.x + threadIdx.x;
  if (e >= E2TOT) return;
  int s, d; const float* row;
  if (e < NE) { s = srcA[e]; d = dstA[e]; row = emb_edge + (size_t)eai[e] * HID; }
  else        { s = e - NE;  d = s;       row = lattr + (size_t)s * HID; }
  float acc[HEADS] = {0.f, 0.f, 0.f, 0.f};
  for (int k = 0; k < HID; ++k) {
    float x = row[k];
#pragma unroll
    for (int h = 0; h < HEADS; ++h) acc[h] += x * vvec[h * HID + k];
  }
#pragma unroll
  for (int h = 0; h < HEADS; ++h) {
    float a = a_s[s * HEADS + h] + a_d[d * HEADS + h] + acc[h];
    a = (a > 0.f) ? a : 0.2f * a;                    // leaky_relu(0.2)
    pre[(size_t)e * HEADS + h] = a;
    atomicMax(&m_u[d * HEADS + h], enc_f(a));
  }
}

// ---------- exp(a - max) and segment sum ----------
__global__ void k_alpha_exp(const int* __restrict__ dstA, const unsigned* __restrict__ m_u,
                            float* __restrict__ pre, float* __restrict__ z) {
  size_t idx = (size_t)blockIdx.x * blockDim.x + threadIdx.x;
  if (idx >= (size_t)E2TOT * HEADS) return;
  int e = (int)(idx >> 2), h = (int)(idx & 3);
  int d = (e < NE) ? dstA[e] : (e - NE);
  float ex = expf(pre[idx] - dec_f(m_u[d * HEADS + h]));
  pre[idx] = ex;
  atomicAdd(&z[d * HEADS + h], ex);
}

// ---------- message scatter: agg[dst,c] += xh[src,c] * alpha ----------
__global__ void k_scatter(const int* __restrict__ srcA, const int* __restrict__ dstA,
                          const float* __restrict__ xh, const float* __restrict__ pre,
                          const float* __restrict__ z, float* __restrict__ agg) {
  size_t idx = (size_t)blockIdx.x * blockDim.x + threadIdx.x;
  if (idx >= (size_t)E2TOT * HID) return;
  int e = (int)(idx >> 7), c = (int)(idx & 127), h = c >> 5;
  int s, d;
  if (e < NE) { s = srcA[e]; d = dstA[e]; } else { s = e - NE; d = s; }
  float al = pre[(size_t)e * HEADS + h] / (z[d * HEADS + h] + 1e-16f);
  atomicAdd(&agg[(size_t)d * HID + c], xh[(size_t)s * HID + c] * al);
}

// ---------- conv bias + BN + ReLU ----------
__global__ void k_bnrelu(const float* __restrict__ agg, const float* __restrict__ conv_b,
                         const float* __restrict__ g, const float* __restrict__ b,
                         float* __restrict__ h) {
  size_t idx = (size_t)blockIdx.x * blockDim.x + threadIdx.x;
  if (idx >= (size_t)NN * HID) return;
  int c = (int)(idx & 127);
  float v = (agg[idx] + conv_b[c]) * (g[c] * INV_BN) + b[c];
  h[idx] = fmaxf(v, 0.f);
}

// ---------- graph pooling (post-ReLU h >= 0 so raw uint max is valid) ----------
__global__ void k_pool(const int* __restrict__ batch, const float* __restrict__ h,
                       float* __restrict__ msum, unsigned* __restrict__ maxp,
                       float* __restrict__ cnt) {
  size_t idx = (size_t)blockIdx.x * blockDim.x + threadIdx.x;
  if (idx >= (size_t)NN * HID) return;
  int n = (int)(idx >> 7), c = (int)(idx & 127);
  int b = batch[n];
  float v = h[idx];
  atomicAdd(&msum[(size_t)b * HID + c], v);
  atomicMax(&maxp[(size_t)b * HID + c], __float_as_uint(v));
  if (c == 0) atomicAdd(&cnt[b], 1.0f);
}

// ---------- build comb = [mean, max, d2, n_feat] ----------
__global__ void k_comb(const float* __restrict__ msum, const float* __restrict__ maxp,
                       const float* __restrict__ cnt, const float* __restrict__ d2,
                       const float* __restrict__ n_feat, float* __restrict__ comb) {
  int idx = blockIdx.x * blockDim.x + threadIdx.x;
  if (idx >= NG * 385) return;
  int g = idx / 385, j = idx % 385;
  float v;
  if (j < 128)       v = msum[(size_t)g * HID + j] / fmaxf(cnt[g], 1.f);
  else if (j < 256)  v = maxp[(size_t)g * HID + (j - 128)];
  else if (j < 384)  v = d2[(size_t)g * HID + (j - 256)];
  else               v = n_feat[g];
  comb[idx] = v;
}

// ---------- small dense layer (head): out = epi(X @ W^T) ----------
__global__ void k_linear(const float* __restrict__ X, const float* __restrict__ W,
                         const float* __restrict__ b, const float* __restrict__ g,
                         const float* __restrict__ beta, float* __restrict__ Out,
                         int M, int K, int O, int relu) {
  int idx = blockIdx.x * blockDim.x + threadIdx.x;
  if (idx >= M * O) return;
  int m = idx / O, o = idx % O;
  const float* x = X + (size_t)m * K;
  const float* w = W + (size_t)o * K;
  float acc = 0.f;
  for (int k = 0; k < K; ++k) acc += x[k] * w[k];
  if (b) acc += b[o];
  if (g) acc = acc * (g[o] * INV_BN) + beta[o];
  if (relu) acc = fmaxf(acc, 0.f);
  Out[idx] = acc;
}

// ==================================================================
extern "C" void kernel_launch(void* const* d_in, const int* in_sizes, int n_in,
                              void* d_out, int out_size, void* d_ws, size_t ws_size,
                              hipStream_t stream) {
  const int*   x_idx   = (const int*)d_in[0];
  const float* x_feat  = (const float*)d_in[1];
  const int*   ei      = (const int*)d_in[2];
  const int*   eai     = (const int*)d_in[3];
  const int*   batch   = (const int*)d_in[4];
  const float* n_feat  = (const float*)d_in[5];
  const float* desc    = (const float*)d_in[6];
  const float* emb_node= (const float*)d_in[7];
  const float* w_nf    = (const float*)d_in[8];
  const float* b_nf    = (const float*)d_in[9];
  const float* emb_edge= (const float*)d_in[10];
  const float* w_lin   = (const float*)d_in[11];
  const float* att_src = (const float*)d_in[12];
  const float* att_dst = (const float*)d_in[13];
  const float* att_edge= (const float*)d_in[14];
  const float* w_edge  = (const float*)d_in[15];
  const float* conv_b  = (const float*)d_in[16];
  const float* bn_g    = (const float*)d_in[17];
  const float* bn_b    = (const float*)d_in[18];
  const float* wd1 = (const float*)d_in[19]; const float* bd1 = (const float*)d_in[20];
  const float* gd1 = (const float*)d_in[21]; const float* betad1 = (const float*)d_in[22];
  const float* wd2 = (const float*)d_in[23]; const float* bd2 = (const float*)d_in[24];
  const float* gd2 = (const float*)d_in[25]; const float* betad2 = (const float*)d_in[26];
  const float* wr0 = (const float*)d_in[27]; const float* br0 = (const float*)d_in[28];
  const float* gr0 = (const float*)d_in[29]; const float* betar0 = (const float*)d_in[30];
  const float* wrh = (const float*)d_in[31]; const float* brh = (const float*)d_in[32];
  const float* grh = (const float*)d_in[33]; const float* betarh = (const float*)d_in[34];
  const float* wo  = (const float*)d_in[35]; const float* bo  = (const float*)d_in[36];

  const int* srcA = ei;
  const int* dstA = ei + NE;

  // ---- workspace carve-up (floats) ----
  float* base = (float*)d_ws;
  size_t off = 0;
  auto alloc = [&](size_t n) { float* p = base + off; off += n; return p; };
  float* h     = alloc((size_t)NN * HID);
  float* xh    = alloc((size_t)NN * HID);
  float* agg   = alloc((size_t)NN * HID);
  float* lattr = alloc((size_t)NN * HID);
  float* deg   = alloc(NN);
  float* a_s   = alloc((size_t)NN * HEADS);
  float* a_d   = alloc((size_t)NN * HEADS);
  unsigned* m_u= (unsigned*)alloc((size_t)NN * HEADS);
  float* zbuf  = alloc((size_t)NN * HEADS);
  float* pre   = alloc((size_t)E2TOT * HEADS);
  float* vvec  = alloc(HEADS * HID);
  float* cnt   = alloc(NG);
  float* msum  = alloc((size_t)NG * HID);
  float* maxp  = alloc((size_t)NG * HID);
  float* d1    = alloc((size_t)NG * 256);
  float* d2b   = alloc((size_t)NG * HID);
  float* comb  = alloc((size_t)NG * 385);
  float* r0    = alloc((size_t)NG * RH);
  float* r1    = alloc((size_t)NG * RH);

  const int B = 256;
  auto NB = [](size_t n, int b) { return (unsigned)((n + b - 1) / b); };

  // 1) node embedding
  k_embed<<<NB((size_t)NN * HID, B), B, 0, stream>>>(x_idx, x_feat, emb_node, w_nf, b_nf, h);

  // 2) self-loop mean edge attrs
  hipMemsetAsync(deg, 0, NN * sizeof(float), stream);
  hipMemsetAsync(lattr, 0, (size_t)NN * HID * sizeof(float), stream);
  k_deg<<<NB(NE, B), B, 0, stream>>>(dstA, deg);
  k_loop_sum<<<NB((size_t)NE * HID, B), B, 0, stream>>>(dstA, eai, emb_edge, lattr);
  k_loop_div<<<NB((size_t)NN * HID, B), B, 0, stream>>>(deg, lattr);

  // 3) GAT layers
  for (int i = 0; i < NLAY; ++i) {
    // xh = h @ w_lin[i]^T  (WMMA fp32, TDM-staged W)
    dim3 gg(NN / 64, HID / 128);
    k_gemm_wmma<<<gg, 128, 0, stream>>>(h, w_lin + (size_t)i * HID * HID, xh,
                                        nullptr, nullptr, nullptr, HID, HID, 0);
    k_attn_nodes<<<NB((size_t)NN * HEADS, B), B, 0, stream>>>(
        xh, att_src + i * HEADS * DHD, att_dst + i * HEADS * DHD, a_s, a_d);
    k_edge_vec<<<NB(HEADS * HID, B), B, 0, stream>>>(
        w_edge + (size_t)i * HID * HID, att_edge + i * HEADS * DHD, vvec);

    hipMemsetAsync(m_u, 0, (size_t)NN * HEADS * sizeof(unsigned), stream);
    hipMemsetAsync(zbuf, 0, (size_t)NN * HEADS * sizeof(float), stream);
    hipMemsetAsync(agg, 0, (size_t)NN * HID * sizeof(float), stream);

    k_alpha_max<<<NB(E2TOT, B), B, 0, stream>>>(srcA, dstA, eai, emb_edge, lattr,
                                                vvec, a_s, a_d, pre, m_u);
    k_alpha_exp<<<NB((size_t)E2TOT * HEADS, B), B, 0, stream>>>(dstA, m_u, pre, zbuf);
    k_scatter<<<NB((size_t)E2TOT * HID, B), B, 0, stream>>>(srcA, dstA, xh, pre, zbuf, agg);
    k_bnrelu<<<NB((size_t)NN * HID, B), B, 0, stream>>>(
        agg, conv_b + i * HID, bn_g + i * HID, bn_b + i * HID, h);
  }

  // 4) pooling
  hipMemsetAsync(cnt, 0, NG * sizeof(float), stream);
  hipMemsetAsync(msum, 0, (size_t)NG * HID * sizeof(float), stream);
  hipMemsetAsync(maxp, 0, (size_t)NG * HID * sizeof(float), stream);
  k_pool<<<NB((size_t)NN * HID, B), B, 0, stream>>>(batch, h, msum, (unsigned*)maxp, cnt);

  // 5) descriptor MLP (WMMA, fused bias+BN+ReLU)
  {
    dim3 g1(NG / 64, 256 / 128);
    k_gemm_wmma<<<g1, 128, 0, stream>>>(desc, wd1, d1, bd1, gd1, betad1, DESC, 256, 1);
    dim3 g2(NG / 64, 128 / 128);
    k_gemm_wmma<<<g2, 128, 0, stream>>>(d1, wd2, d2b, bd2, gd2, betad2, 256, 128, 1);
  }

  // 6) comb + regression head
  k_comb<<<NB((size_t)NG * 385, B), B, 0, stream>>>(msum, maxp, cnt, d2b, n_feat, comb);
  k_linear<<<NB((size_t)NG * RH, B), B, 0, stream>>>(comb, wr0, br0, gr0, betar0, r0,
                                                     NG, 385, RH, 1);
  float* rin = r0; float* rout = r1;
  for (int i = 0; i < NLAY; ++i) {
    k_linear<<<NB((size_t)NG * RH, B), B, 0, stream>>>(
        rin, wrh + (size_t)i * RH * RH, brh + i * RH, grh + i * RH, betarh + i * RH,
        rout, NG, RH, RH, 1);
    float* t = rin; rin = rout; rout = t;
  }
  k_linear<<<NB(NG, B), B, 0, stream>>>(rin, wo, bo, nullptr, nullptr, (float*)d_out,
                                        NG, RH, 1, 0);
}